// PointNetFeaturePropagation_9414568312916
// MI455X (gfx1250) — compile-verified
//
#include <hip/hip_runtime.h>

// Problem constants (match reference)
#define B_ 8
#define N_ 8192
#define S_ 2048
#define D_ 256
#define EPS_ 1e-8f
#define FLT_MAX_ 3.402823466e+38f

typedef __attribute__((ext_vector_type(2))) float v2f;
typedef __attribute__((ext_vector_type(8))) float v8f;

__device__ __forceinline__ unsigned umin_(unsigned a, unsigned b) { return a < b ? a : b; }
__device__ __forceinline__ unsigned umax_(unsigned a, unsigned b) { return a > b ? a : b; }

// Branchless sorted insert of `key` into ascending triple (k0 <= k1 <= k2):
// 5 v_min_u32/v_max_u32 ops, dependency depth 3.
#define INS3(key, k0, k1, k2)                                   \
  do {                                                          \
    unsigned _t0 = umin_((key), (k0));                          \
    unsigned _h0 = umax_((key), (k0));                          \
    (k0) = _t0;                                                 \
    unsigned _t1 = umin_(_h0, (k1));                            \
    unsigned _h1 = umax_(_h0, (k1));                            \
    (k1) = _t1;                                                 \
    (k2) = umin_(_h1, (k2));                                    \
  } while (0)

// One workgroup = 256 threads = 8 waves; each wave owns 16 queries.
// grid = (N/128, B)
__global__ __launch_bounds__(256) void fp_interp_kernel(
    const float* __restrict__ xyz1,    // [B,3,N]
    const float* __restrict__ xyz2,    // [B,3,S]
    const float* __restrict__ points2, // [B,S,D]
    float* __restrict__ out)           // [B,N,D]
{
  // Per-source A-operand staging: (-2x, -2y, -2z, |p|^2), 32KB LDS.
  __shared__ float4 sA[S_];

  const int b    = blockIdx.y;
  const int tid  = threadIdx.x;
  const int lane = tid & 31;
  const int wv   = tid >> 5;
  const bool lo  = (lane < 16);

  // ---- Stage source points into LDS in WMMA A-operand-friendly form ----
  const float* x2 = xyz2 + (size_t)b * 3 * S_;
  for (int s = tid; s < S_; s += 256) {
    float px = x2[s];
    float py = x2[S_ + s];
    float pz = x2[2 * S_ + s];
    sA[s] = make_float4(-2.f * px, -2.f * py, -2.f * pz,
                        px * px + py * py + pz * pz);
  }
  __syncthreads();

  // ---- Per-lane query (B operand + C init) ----
  const int n0 = blockIdx.x * 128 + wv * 16;      // first query of this wave
  const int n  = n0 + (lane & 15);                // this lane's query
  const float* x1 = xyz1 + (size_t)b * 3 * N_;
  const float qx = x1[n];
  const float qy = x1[N_ + n];
  const float qz = x1[2 * N_ + n];
  const float qq = qx * qx + qy * qy + qz * qz;

  // B (4x16, KxN): V0 = {K=0 | K=2}, V1 = {K=1 | K=3} across lane halves.
  v2f bv;
  bv.x = lo ? qx : qz;
  bv.y = lo ? qy : 1.0f;

  // C = |q|^2 broadcast (lane's column is fixed across all 8 C VGPRs).
  v8f ci;
#pragma unroll
  for (int i = 0; i < 8; ++i) ci[i] = qq;

  // A (16x4, MxK): lane<16 reads (K0,K1)=(-2x,-2y), lane>=16 reads (K2,K3)=(-2z,|p|^2)
  const float2* aptr = (const float2*)((const char*)sA +
                       (size_t)(((lane & 15) * 16) + (lo ? 0 : 8)));
  const unsigned hi8 = lo ? 0u : 8u;  // source-row offset of this half-lane

  // Two independent top-3 chains (even/odd candidates) to double ILP of the
  // serial min/max insertion dependency chain.
  unsigned kA0 = 0xFFFFFFFFu, kA1 = 0xFFFFFFFFu, kA2 = 0xFFFFFFFFu;
  unsigned kB0 = 0xFFFFFFFFu, kB1 = 0xFFFFFFFFu, kB2 = 0xFFFFFFFFu;

  // ---- Main loop: 128 tiles of 16 sources; 1 WMMA = full 16x16 dist tile ----
#pragma unroll 4
  for (int t = 0; t < S_ / 16; ++t) {
    float2 a2 = aptr[t * 32];   // ds_load_b64, conflict-free
    v2f av; av.x = a2.x; av.y = a2.y;

    // D[m][n] = (-2 p_m) . q_n + |p_m|^2 + |q_n|^2  == squared distance
    v8f dv = __builtin_amdgcn_wmma_f32_16x16x4_f32(
        false, av, false, bv, (short)0, ci, false, false);

    // Index fields are bit-disjoint: t*16 (bits>=4) | hi8 (bit 3) | k (bits 0-2)
    const unsigned sb = ((unsigned)t * 16u) | hi8;
#pragma unroll
    for (int k = 0; k < 8; ++k) {
      // clamp to [0, +inf) in ONE v_med3 (fused formula can go slightly neg)
      float d = __builtin_amdgcn_fmed3f(dv[k], 0.0f, FLT_MAX_);
      // distance-major key; low 11 mantissa bits carry the source index.
      // All three OR fields disjoint -> v_and + v_or3 (k is inline const).
      unsigned key = (__float_as_uint(d) & 0xFFFFF800u) | sb | (unsigned)k;
      if (k & 1) { INS3(key, kB0, kB1, kB2); }
      else       { INS3(key, kA0, kA1, kA2); }
    }
  }

  // ---- Merge odd chain into even chain ----
  unsigned k0 = kA0, k1 = kA1, k2 = kA2;
  INS3(kB0, k0, k1, k2);
  INS3(kB1, k0, k1, k2);
  INS3(kB2, k0, k1, k2);

  // ---- Merge the two half-lane top-3 sets (lane L <-> L^16) ----
  {
    unsigned o0 = (unsigned)__shfl_xor((int)k0, 16, 32);
    unsigned o1 = (unsigned)__shfl_xor((int)k1, 16, 32);
    unsigned o2 = (unsigned)__shfl_xor((int)k2, 16, 32);
    INS3(o0, k0, k1, k2);
    INS3(o1, k0, k1, k2);
    INS3(o2, k0, k1, k2);
  }

  // ---- Unpack + inverse-distance weights ----
  const float d0 = __uint_as_float(k0 & 0xFFFFF800u);
  const float d1 = __uint_as_float(k1 & 0xFFFFF800u);
  const float d2 = __uint_as_float(k2 & 0xFFFFF800u);
  const int i0 = (int)(k0 & 0x7FFu);
  const int i1 = (int)(k1 & 0x7FFu);
  const int i2 = (int)(k2 & 0x7FFu);
  const float r0 = 1.0f / (d0 + EPS_);
  const float r1 = 1.0f / (d1 + EPS_);
  const float r2 = 1.0f / (d2 + EPS_);
  const float inv = 1.0f / (r0 + r1 + r2);
  const float w0 = r0 * inv, w1 = r1 * inv, w2 = r2 * inv;

  // ---- Phase 2: gather + weighted sum, whole wave per query ----
  const float* p2b = points2 + (size_t)b * S_ * D_;
  float* outw = out + ((size_t)b * N_ + (size_t)n0) * D_;

  for (int q = 0; q < 16; ++q) {
    const int   j0 = __shfl(i0, q, 32);
    const int   j1 = __shfl(i1, q, 32);
    const int   j2 = __shfl(i2, q, 32);
    const float a0 = __shfl(w0, q, 32);
    const float a1 = __shfl(w1, q, 32);
    const float a2 = __shfl(w2, q, 32);

    const float4* r0p = (const float4*)(p2b + (size_t)j0 * D_);
    const float4* r1p = (const float4*)(p2b + (size_t)j1 * D_);
    const float4* r2p = (const float4*)(p2b + (size_t)j2 * D_);
    float4* ov = (float4*)(outw + (size_t)q * D_);

#pragma unroll
    for (int c = 0; c < 2; ++c) {
      const int idx = lane + c * 32;      // 64 float4 per row
      float4 va = r0p[idx];
      float4 vb = r1p[idx];
      float4 vc = r2p[idx];
      float4 acc;
      acc.x = a0 * va.x + a1 * vb.x + a2 * vc.x;
      acc.y = a0 * va.y + a1 * vb.y + a2 * vc.y;
      acc.z = a0 * va.z + a1 * vb.z + a2 * vc.z;
      acc.w = a0 * va.w + a1 * vb.w + a2 * vc.w;
      ov[idx] = acc;
    }
  }
}

extern "C" void kernel_launch(void* const* d_in, const int* in_sizes, int n_in,
                              void* d_out, int out_size, void* d_ws, size_t ws_size,
                              hipStream_t stream) {
  const float* xyz1    = (const float*)d_in[0];  // [B,3,N]
  const float* xyz2    = (const float*)d_in[1];  // [B,3,S]
  // d_in[2] = points1 — unused by the reference computation
  const float* points2 = (const float*)d_in[3];  // [B,S,D]
  float* out = (float*)d_out;                    // [B,N,D]

  dim3 grid(N_ / 128, B_);
  dim3 block(256);
  hipLaunchKernelGGL(fp_interp_kernel, grid, block, 0, stream,
                     xyz1, xyz2, points2, out);
}